// RGAT_64536178589820
// MI455X (gfx1250) — compile-verified
//
#include <hip/hip_runtime.h>

typedef float v2f __attribute__((ext_vector_type(2)));
typedef float v8f __attribute__((ext_vector_type(8)));

#define ND 10000
#define NT 10000
#define NE 150000
#define HH 2
#define FO 32

// ---------------- utility ----------------
__global__ void fill_f32(float* __restrict__ p, float v, int n) {
  int i = blockIdx.x * blockDim.x + threadIdx.x;
  if (i < n) p[i] = v;
}

// ---------------- GEMM: C[M x 64] = A[M x K] @ W[64 x K]^T  (V_WMMA_F32_16X16X4_F32)
// One wave -> one 16-row M tile, all 64 output columns (4 accumulators).
// A frag (16x4): lane L holds M = L&15, K-pair 2*(L>>4)+{0,1}.
// B frag (4x16): lane L holds N = L&15, K-pair 2*(L>>4)+{0,1}.  B[k][n] = W[n][k].
// C/D (16x16):   VGPR v, lane L -> M = v + 8*(L>>4), N = L&15.
__global__ void __launch_bounds__(256)
gemm_wmma_f32(const float* __restrict__ A, const float* __restrict__ W,
              float* __restrict__ C, int M, int K) {
  int wave = (blockIdx.x * blockDim.x + threadIdx.x) >> 5;
  int lane = threadIdx.x & 31;
  int m0 = wave * 16;
  if (m0 >= M) return;
  int l15 = lane & 15;
  int khi = (lane >> 4) << 1;  // 0 or 2

  v8f acc0 = {0.f, 0.f, 0.f, 0.f, 0.f, 0.f, 0.f, 0.f};
  v8f acc1 = acc0, acc2 = acc0, acc3 = acc0;

  const float* arow = A + (size_t)(m0 + l15) * K + khi;
  const float* wrow = W + (size_t)l15 * K + khi;
  for (int k = 0; k < K; k += 4) {
    v2f a;
    a.x = arow[k];
    a.y = arow[k + 1];
    v2f b0, b1, b2, b3;
    b0.x = wrow[k];              b0.y = wrow[k + 1];
    b1.x = wrow[k + 16 * K];     b1.y = wrow[k + 16 * K + 1];
    b2.x = wrow[k + 32 * K];     b2.y = wrow[k + 32 * K + 1];
    b3.x = wrow[k + 48 * K];     b3.y = wrow[k + 48 * K + 1];
    acc0 = __builtin_amdgcn_wmma_f32_16x16x4_f32(false, a, false, b0, (short)0, acc0, false, false);
    acc1 = __builtin_amdgcn_wmma_f32_16x16x4_f32(false, a, false, b1, (short)0, acc1, false, false);
    acc2 = __builtin_amdgcn_wmma_f32_16x16x4_f32(false, a, false, b2, (short)0, acc2, false, false);
    acc3 = __builtin_amdgcn_wmma_f32_16x16x4_f32(false, a, false, b3, (short)0, acc3, false, false);
  }

  int mrow = m0 + ((lane >> 4) << 3);
#pragma unroll
  for (int v = 0; v < 8; ++v) {
    float* crow = C + (size_t)(mrow + v) * 64 + l15;
    crow[0]  = acc0[v];
    crow[16] = acc1[v];
    crow[32] = acc2[v];
    crow[48] = acc3[v];
  }
}

// ---------------- el/er: out[n*PH+ph] = dot(z[n,ph,:], a[h,:]) ----------------
__global__ void attn_dot(const float* __restrict__ z, const float* __restrict__ avec,
                         float* __restrict__ out, int total /* = N*PH */) {
  int i = blockIdx.x * blockDim.x + threadIdx.x;
  if (i >= total) return;
  int h = i % HH;  // ph = p*HH + h, PH divisible by HH
  const float* zr = z + (size_t)i * FO;
  const float* ar = avec + h * FO;
  float s = 0.f;
#pragma unroll
  for (int f = 0; f < FO; ++f) s += zr[f] * ar[f];
  out[i] = s;
}

__device__ __forceinline__ float leaky(float v) { return v > 0.f ? v : 0.2f * v; }

// ---------------- edge softmax pass 1: segment max ----------------
__global__ void edge_max(const int* __restrict__ src, const int* __restrict__ dst,
                         const float* __restrict__ el, const float* __restrict__ er,
                         float* __restrict__ m, int PH) {
  int t = blockIdx.x * blockDim.x + threadIdx.x;
  int e = t / PH, ph = t - e * PH;
  if (e >= NE) return;
  int s = src[e], d = dst[e];
  float v = leaky(el[s * PH + ph] + er[d * PH + ph]);
  atomicMax(&m[d * PH + ph], v);
}

// ---------------- edge softmax pass 2: exp + segment sum ----------------
__global__ void edge_expsum(const int* __restrict__ src, const int* __restrict__ dst,
                            const float* __restrict__ el, const float* __restrict__ er,
                            const float* __restrict__ m, float* __restrict__ ex,
                            float* __restrict__ ssum, int PH) {
  int t = blockIdx.x * blockDim.x + threadIdx.x;
  int e = t / PH, ph = t - e * PH;
  if (e >= NE) return;
  int s = src[e], d = dst[e];
  float v = leaky(el[s * PH + ph] + er[d * PH + ph]);
  float x = expf(v - m[d * PH + ph]);
  ex[t] = x;
  atomicAdd(&ssum[d * PH + ph], x);
}

// ---------------- edge message scatter: acc[dst,ph,:] += alpha * z_src[src,ph,:]
__global__ void edge_scatter(const int* __restrict__ src, const int* __restrict__ dst,
                             const float* __restrict__ ex, const float* __restrict__ ssum,
                             const float* __restrict__ zsrc, float* __restrict__ acc,
                             int PH) {
  int t = blockIdx.x * blockDim.x + threadIdx.x;
  int e = t / PH, ph = t - e * PH;
  if (e >= NE) return;
  int s = src[e], d = dst[e];
  float alpha = ex[t] / ssum[d * PH + ph];
  const float4* zp = (const float4*)(zsrc + ((size_t)s * PH + ph) * FO);
  float* ap = acc + ((size_t)d * PH + ph) * FO;
#pragma unroll
  for (int q = 0; q < FO / 4; ++q) {
    float4 z = zp[q];
    atomicAdd(&ap[q * 4 + 0], alpha * z.x);
    atomicAdd(&ap[q * 4 + 1], alpha * z.y);
    atomicAdd(&ap[q * 4 + 2], alpha * z.z);
    atomicAdd(&ap[q * 4 + 3], alpha * z.w);
  }
}

// ---------------- finalize: relu(acc + b1 + b2) for layers 1,2 ----------------
__global__ void finalize_relu(const float* __restrict__ acc, const float* __restrict__ b1,
                              const float* __restrict__ b2, float* __restrict__ hout,
                              int total /* = N*PH*FO */) {
  int i = blockIdx.x * blockDim.x + threadIdx.x;
  if (i >= total) return;
  int f = i % FO;
  int h = (i / FO) % HH;
  float v = acc[i] + b1[h * FO + f] + b2[h * FO + f];
  hout[i] = v > 0.f ? v : 0.f;
}

// ---------------- finalize layer 3: mean over 8 (prefix x head) groups ----------------
__global__ void finalize_mean(const float* __restrict__ acc, const float* __restrict__ b1,
                              const float* __restrict__ b2, float* __restrict__ out,
                              int N) {
  int i = blockIdx.x * blockDim.x + threadIdx.x;
  if (i >= N * FO) return;
  int n = i / FO, f = i - n * FO;
  float s = 0.f;
#pragma unroll
  for (int g = 0; g < 8; ++g) {
    int h = g % HH;
    s += acc[(size_t)n * 8 * FO + g * FO + f] + b1[h * FO + f] + b2[h * FO + f];
  }
  out[i] = s * 0.125f;
}

// ---------------- host orchestration ----------------
static inline int cdiv(long long a, long long b) { return (int)((a + b - 1) / b); }

extern "C" void kernel_launch(void* const* d_in, const int* in_sizes, int n_in,
                              void* d_out, int out_size, void* d_ws, size_t ws_size,
                              hipStream_t stream) {
  (void)in_sizes; (void)n_in; (void)out_size; (void)ws_size;

  const float* x_d = (const float*)d_in[0];
  const float* x_t = (const float*)d_in[1];
  const int* esrc[4];
  const int* edst[4];
  for (int r = 0; r < 4; ++r) {
    esrc[r] = (const int*)d_in[2 + 2 * r];
    edst[r] = (const int*)d_in[3 + 2 * r];
  }
  const float *Wp[3][4], *AL[3][4], *AR[3][4], *Bp[3][4];
  int pi = 10;
  for (int l = 0; l < 3; ++l)
    for (int r = 0; r < 4; ++r) {
      Wp[l][r] = (const float*)d_in[pi++];
      AL[l][r] = (const float*)d_in[pi++];
      AR[l][r] = (const float*)d_in[pi++];
      Bp[l][r] = (const float*)d_in[pi++];
    }

  // workspace carve-up (floats)
  float* ws = (float*)d_ws;
  size_t off = 0;
  auto alloc = [&](size_t n) { float* p = ws + off; off += n; return p; };
  float* hbuf[2][2];                       // [type][pingpong], max [10000,4,32]
  hbuf[0][0] = alloc((size_t)ND * 4 * FO);
  hbuf[0][1] = alloc((size_t)ND * 4 * FO);
  hbuf[1][0] = alloc((size_t)NT * 4 * FO);
  hbuf[1][1] = alloc((size_t)NT * 4 * FO);
  float* accb[2];
  accb[0] = alloc((size_t)ND * 8 * FO);    // max PH_out = 8
  accb[1] = alloc((size_t)NT * 8 * FO);
  float* zA = alloc((size_t)ND * 8 * FO);  // z_src, max [10000, 8, 32]
  float* zB = alloc((size_t)NT * 8 * FO);  // z_dst
  float* elb = alloc((size_t)ND * 8);
  float* erb = alloc((size_t)NT * 8);
  float* mb  = alloc((size_t)ND * 8);
  float* sb  = alloc((size_t)ND * 8);
  float* exb = alloc((size_t)NE * 8);

  const int stype[4] = {0, 0, 1, 1};       // dd, dt, td, tt
  const int dtype[4] = {0, 1, 0, 1};
  const int Nn[2] = {ND, NT};
  const int Pin[3] = {1, 2, 4};
  const int Kfi[3] = {256, 32, 32};

  const float* hin[2] = {x_d, x_t};
  float* out_d = (float*)d_out;
  float* out_t = out_d + (size_t)ND * FO;

  const int TB = 256;
  for (int l = 0; l < 3; ++l) {
    int P = Pin[l], K = Kfi[l], PH = P * HH;
    // zero the per-type accumulators
    for (int ty = 0; ty < 2; ++ty) {
      int n = Nn[ty] * PH * FO;
      hipLaunchKernelGGL(fill_f32, dim3(cdiv(n, TB)), dim3(TB), 0, stream, accb[ty], 0.f, n);
    }
    for (int r = 0; r < 4; ++r) {
      int st = stype[r], dt = dtype[r];
      int Ms = Nn[st] * P, Md = Nn[dt] * P;
      // z_src = h[st] @ W^T
      {
        int th = (Ms / 16) * 32;
        hipLaunchKernelGGL(gemm_wmma_f32, dim3(cdiv(th, TB)), dim3(TB), 0, stream,
                           hin[st], Wp[l][r], zA, Ms, K);
      }
      float* zdst = zA;
      if (st != dt) {
        int th = (Md / 16) * 32;
        hipLaunchKernelGGL(gemm_wmma_f32, dim3(cdiv(th, TB)), dim3(TB), 0, stream,
                           hin[dt], Wp[l][r], zB, Md, K);
        zdst = zB;
      }
      // attention logits per node
      {
        int n = Nn[st] * PH;
        hipLaunchKernelGGL(attn_dot, dim3(cdiv(n, TB)), dim3(TB), 0, stream, zA, AL[l][r], elb, n);
      }
      {
        int n = Nn[dt] * PH;
        hipLaunchKernelGGL(attn_dot, dim3(cdiv(n, TB)), dim3(TB), 0, stream, zdst, AR[l][r], erb, n);
      }
      // segment max / sum init
      {
        int n = Nn[dt] * PH;
        hipLaunchKernelGGL(fill_f32, dim3(cdiv(n, TB)), dim3(TB), 0, stream, mb, -INFINITY, n);
        hipLaunchKernelGGL(fill_f32, dim3(cdiv(n, TB)), dim3(TB), 0, stream, sb, 0.f, n);
      }
      int en = NE * PH;
      hipLaunchKernelGGL(edge_max, dim3(cdiv(en, TB)), dim3(TB), 0, stream,
                         esrc[r], edst[r], elb, erb, mb, PH);
      hipLaunchKernelGGL(edge_expsum, dim3(cdiv(en, TB)), dim3(TB), 0, stream,
                         esrc[r], edst[r], elb, erb, mb, exb, sb, PH);
      hipLaunchKernelGGL(edge_scatter, dim3(cdiv(en, TB)), dim3(TB), 0, stream,
                         esrc[r], edst[r], exb, sb, zA, accb[dt], PH);
    }
    // finalize: acc_d gets biases of relations dd(r0)+td(r2); acc_t gets dt(r1)+tt(r3)
    if (l < 2) {
      float* hnd = hbuf[0][l & 1];
      float* hnt = hbuf[1][l & 1];
      int n0 = ND * PH * FO, n1 = NT * PH * FO;
      hipLaunchKernelGGL(finalize_relu, dim3(cdiv(n0, TB)), dim3(TB), 0, stream,
                         accb[0], Bp[l][0], Bp[l][2], hnd, n0);
      hipLaunchKernelGGL(finalize_relu, dim3(cdiv(n1, TB)), dim3(TB), 0, stream,
                         accb[1], Bp[l][1], Bp[l][3], hnt, n1);
      hin[0] = hnd;
      hin[1] = hnt;
    } else {
      hipLaunchKernelGGL(finalize_mean, dim3(cdiv(ND * FO, TB)), dim3(TB), 0, stream,
                         accb[0], Bp[2][0], Bp[2][2], out_d, ND);
      hipLaunchKernelGGL(finalize_mean, dim3(cdiv(NT * FO, TB)), dim3(TB), 0, stream,
                         accb[1], Bp[2][1], Bp[2][3], out_t, NT);
    }
  }
}